// GenTargets_5669356833377
// MI455X (gfx1250) — compile-verified
//
#include <hip/hip_runtime.h>

// FCOS GenTargets for MI455X (gfx1250, wave32).
// Only gt_boxes (d_in[15]) and classes (d_in[16]) are consumed; the 15 logits
// tensors only define shapes, which are compile-time constants here.
//
// CDNA5-specific path: async global->LDS broadcast of the per-batch GT table
// (GLOBAL_LOAD_ASYNC_TO_LDS_B32, tracked by ASYNCcnt / s_wait_asynccnt).

#define AS1 __attribute__((address_space(1)))
#define AS3 __attribute__((address_space(3)))

#if defined(__AMDGCN__) && __has_builtin(__builtin_amdgcn_global_load_async_to_lds_b32)
#define HAVE_ASYNC_LDS 1
#endif

__device__ __forceinline__ void async_cp_b32(const void* g, void* l) {
#ifdef HAVE_ASYNC_LDS
  __builtin_amdgcn_global_load_async_to_lds_b32(
      (AS1 int*)const_cast<void*>(g), (AS3 int*)l, /*offset=*/0, /*cpol=*/0);
#else
  *(unsigned*)l = *(const unsigned*)g;
#endif
}

__device__ __forceinline__ void wait_async_lds() {
#ifdef HAVE_ASYNC_LDS
#if __has_builtin(__builtin_amdgcn_s_wait_asynccnt)
  __builtin_amdgcn_s_wait_asynccnt(0);
#else
  asm volatile("s_wait_asynccnt 0" ::: "memory");
#endif
#endif
}

#define BATCH    8
#define MBOX     100
#define TOTAL_HW 17064   // 12800 + 3200 + 800 + 208 + 56
#define BIGAREA  99999999.0f

__global__ __launch_bounds__(256) void fcos_gen_targets_kernel(
    const float* __restrict__ gt,   // [B, M, 4]  x1 y1 x2 y2
    const int*   __restrict__ cls,  // [B, M]
    float*       __restrict__ out)  // cls[B*HW] | cnt[B*HW] | reg[B*HW*4]
{
  __shared__ alignas(16) unsigned raw[MBOX * 4 + MBOX];  // 500 dwords
  __shared__ alignas(16) float    tbl[MBOX][8];          // x1 y1 x2 y2 cx cy clsf pad

  const int b = blockIdx.y;
  const unsigned* gb = (const unsigned*)(gt + (size_t)b * MBOX * 4);
  const unsigned* cb = (const unsigned*)(cls + (size_t)b * MBOX);

  // ---- async broadcast of this batch's GT data into LDS --------------------
  for (int i = threadIdx.x; i < MBOX * 4; i += blockDim.x)
    async_cp_b32(gb + i, &raw[i]);
  for (int i = threadIdx.x; i < MBOX; i += blockDim.x)
    async_cp_b32(cb + i, &raw[MBOX * 4 + i]);
  wait_async_lds();
  __syncthreads();

  // ---- per-box derived table (center + class as float) ---------------------
  if (threadIdx.x < MBOX) {
    const float* bx = (const float*)&raw[threadIdx.x * 4];
    const float x1 = bx[0], y1 = bx[1], x2 = bx[2], y2 = bx[3];
    tbl[threadIdx.x][0] = x1;
    tbl[threadIdx.x][1] = y1;
    tbl[threadIdx.x][2] = x2;
    tbl[threadIdx.x][3] = y2;
    tbl[threadIdx.x][4] = 0.5f * (x1 + x2);
    tbl[threadIdx.x][5] = 0.5f * (y1 + y2);
    tbl[threadIdx.x][6] = (float)((int)raw[MBOX * 4 + threadIdx.x]);
    tbl[threadIdx.x][7] = 0.0f;
  }
  __syncthreads();

  const int hw = blockIdx.x * blockDim.x + threadIdx.x;
  if (hw >= TOTAL_HW) return;

  // ---- level decode (all widths/strides are powers of two) -----------------
  int base, wsh, ssh;
  float lim0, lim1;
  if (hw < 12800)      { base = 0;     wsh = 7; ssh = 3; lim0 = -1.0f;  lim1 = 64.0f;   }
  else if (hw < 16000) { base = 12800; wsh = 6; ssh = 4; lim0 = 64.0f;  lim1 = 128.0f;  }
  else if (hw < 16800) { base = 16000; wsh = 5; ssh = 5; lim0 = 128.0f; lim1 = 256.0f;  }
  else if (hw < 17008) { base = 16800; wsh = 4; ssh = 6; lim0 = 256.0f; lim1 = 512.0f;  }
  else                 { base = 17008; wsh = 3; ssh = 7; lim0 = 512.0f; lim1 = BIGAREA; }

  const int   local  = hw - base;
  const float stride = (float)(1 << ssh);
  const float x      = (float)(local & ((1 << wsh) - 1)) * stride + 0.5f * stride;
  const float y      = (float)(local >> wsh) * stride + 0.5f * stride;
  const float radiu  = 1.5f * stride;

  // ---- masked-area argmin over 100 boxes (first-index tiebreak) ------------
  float best = 3.0e38f;
  float bl = -1.0f, bt = -1.0f, br = -1.0f, bbo = -1.0f, bcls = 0.0f;
  bool  pos2 = false;

#pragma unroll 2
  for (int m = 0; m < MBOX; ++m) {
    const float4 q0 = *(const float4*)&tbl[m][0];   // x1 y1 x2 y2
    const float4 q1 = *(const float4*)&tbl[m][4];   // cx cy clsf pad
    const float l = x - q0.x;
    const float t = y - q0.y;
    const float r = q0.z - x;
    const float d = q0.w - y;
    const float omin = fminf(fminf(l, t), fminf(r, d));
    const float omax = fmaxf(fmaxf(l, t), fmaxf(r, d));
    const float coff = fmaxf(fabsf(x - q1.x), fabsf(y - q1.y));
    const bool  pos  = (omin > 0.0f) && (omax > lim0) && (omax <= lim1) && (coff < radiu);
    const float area = (l + r) * (t + d);
    const float am   = pos ? area : BIGAREA;
    pos2 = pos2 || pos;
    if (am < best) { best = am; bl = l; bt = t; br = r; bbo = d; bcls = q1.z; }
  }

  // ---- finalize ------------------------------------------------------------
  float ocls = 0.0f, ocnt = -1.0f;
  float4 oreg = make_float4(-1.0f, -1.0f, -1.0f, -1.0f);
  if (pos2) {
    ocls = bcls;
    const float lrmin = fminf(bl, br), lrmax = fmaxf(bl, br);
    const float tbmin = fminf(bt, bbo), tbmax = fmaxf(bt, bbo);
    ocnt = sqrtf((lrmin * tbmin) / (lrmax * tbmax + 1e-10f));
    oreg = make_float4(bl, bt, br, bbo);
  }

  const int idx = b * TOTAL_HW + hw;
  out[idx] = ocls;                                        // cls targets
  out[BATCH * TOTAL_HW + idx] = ocnt;                     // cnt targets
  *(float4*)(out + 2 * BATCH * TOTAL_HW + (size_t)idx * 4) = oreg;  // reg targets
}

extern "C" void kernel_launch(void* const* d_in, const int* in_sizes, int n_in,
                              void* d_out, int out_size, void* d_ws, size_t ws_size,
                              hipStream_t stream) {
  (void)in_sizes; (void)n_in; (void)out_size; (void)d_ws; (void)ws_size;
  // setup_inputs order: (cls,cnt,reg) x 5 levels, then gt_boxes, classes.
  const float* gt  = (const float*)d_in[15];
  const int*   cls = (const int*)d_in[16];
  dim3 grid((TOTAL_HW + 255) / 256, BATCH, 1);
  fcos_gen_targets_kernel<<<grid, dim3(256, 1, 1), 0, stream>>>(gt, cls, (float*)d_out);
}